// Chi2Loss_22874995818943
// MI455X (gfx1250) — compile-verified
//
#include <hip/hip_runtime.h>

typedef float v2f __attribute__((ext_vector_type(2)));
typedef float v8f __attribute__((ext_vector_type(8)));

#define THREADS 256

// ---------------------------------------------------------------------------
// Kernel 1: one 256-thread block (8 wave32) per sample. Streams the three
// f32 windows with coalesced b32 loads, 4x unrolled for memory-level
// parallelism, accumulates (x-y)^2 * w in f32, block-reduces, writes
// per_sample[b] = sum/len into workspace. Deterministic (no atomics).
// ---------------------------------------------------------------------------
__global__ __launch_bounds__(THREADS) void chi2_per_sample(
    const float* __restrict__ fluct,
    const float* __restrict__ ivar,
    const float* __restrict__ outp,
    const long long* __restrict__ ov,   // overlap_index, int64 [B][4]
    float* __restrict__ per_sample,
    int L)
{
    const int b   = blockIdx.x;
    const int tid = threadIdx.x;

    const long long s_in  = ov[4LL * b + 0];
    const long long e_in  = ov[4LL * b + 1];
    const long long s_out = ov[4LL * b + 2];
    const int len = (int)(e_in - s_in);   // in [1024, 3072] by construction

    const float* __restrict__ xp = fluct + (size_t)b * (size_t)L + (size_t)s_in;
    const float* __restrict__ wp = ivar  + (size_t)b * (size_t)L + (size_t)s_in;
    const float* __restrict__ yp = outp  + (size_t)b * (size_t)L + (size_t)s_out;

    float acc = 0.0f;
    int j = tid;
    // 4x unrolled main loop: 12 loads in flight per thread
    for (; j + 3 * THREADS < len; j += 4 * THREADS) {
        float x0 = xp[j];             float x1 = xp[j + THREADS];
        float x2 = xp[j + 2*THREADS]; float x3 = xp[j + 3*THREADS];
        float w0 = wp[j];             float w1 = wp[j + THREADS];
        float w2 = wp[j + 2*THREADS]; float w3 = wp[j + 3*THREADS];
        float y0 = yp[j];             float y1 = yp[j + THREADS];
        float y2 = yp[j + 2*THREADS]; float y3 = yp[j + 3*THREADS];
        float d0 = x0 - y0; acc = fmaf(d0 * d0, w0, acc);
        float d1 = x1 - y1; acc = fmaf(d1 * d1, w1, acc);
        float d2 = x2 - y2; acc = fmaf(d2 * d2, w2, acc);
        float d3 = x3 - y3; acc = fmaf(d3 * d3, w3, acc);
    }
    for (; j < len; j += THREADS) {
        float x = xp[j], w = wp[j], y = yp[j];
        float d = x - y;
        acc = fmaf(d * d, w, acc);
    }

    // wave32 reduce
    #pragma unroll
    for (int m = 16; m > 0; m >>= 1) acc += __shfl_xor(acc, m, 32);

    __shared__ float red[THREADS / 32];
    const int wave = tid >> 5;
    if ((tid & 31) == 0) red[wave] = acc;
    __syncthreads();
    if (tid == 0) {
        float t = 0.0f;
        #pragma unroll
        for (int i = 0; i < THREADS / 32; ++i) t += red[i];
        per_sample[b] = t / (float)len;
    }
}

// ---------------------------------------------------------------------------
// Kernel 2: single wave32. Sums n f32 values with V_WMMA_F32_16X16X4_F32:
// A = ones (16x4), each WMMA folds a 64-value B tile into the f32 C/D
// accumulator: D[m][n] = sum_k B[k][n] + C[m][n] (rows replicated).
// Exact f32 accumulation — no precision loss. Writes mean to out[0].
// ---------------------------------------------------------------------------
__global__ __launch_bounds__(32) void mean_reduce_wmma(
    const float* __restrict__ vals, float* __restrict__ out, int n)
{
    const int lane = threadIdx.x;   // 0..31, single wave, EXEC all ones

    v8f acc = {};
    v2f ones; ones.x = 1.0f; ones.y = 1.0f;

    const int nfull = (n / 64) * 64;
    for (int i = 0; i < nfull; i += 64) {
        v2f bm;
        bm.x = vals[i + lane];
        bm.y = vals[i + lane + 32];
        // D = A(ones) x B + C : adds all 64 tile values into the column sums
        acc = __builtin_amdgcn_wmma_f32_16x16x4_f32(
            /*neg_a=*/false, ones, /*neg_b=*/false, bm,
            /*c_mod=*/(short)0, acc, /*reuse_a=*/false, /*reuse_b=*/false);
    }

    // D row 0 lives in acc[0] lanes 0..15 (lanes 16..31 replicate row 8)
    float r = (lane < 16) ? acc[0] : 0.0f;
    if (lane < 16) {
        for (int i = nfull + lane; i < n; i += 16) r += vals[i];  // tail
    }
    // sum the 16 column sums (xor-reduce within the 16-lane group)
    #pragma unroll
    for (int m = 8; m > 0; m >>= 1) r += __shfl_xor(r, m, 32);

    if (lane == 0) out[0] = r / (float)n;
}

// ---------------------------------------------------------------------------
extern "C" void kernel_launch(void* const* d_in, const int* in_sizes, int n_in,
                              void* d_out, int out_size, void* d_ws, size_t ws_size,
                              hipStream_t stream) {
    const float*     fl = (const float*)d_in[0];      // fluctuate [B,1,L]
    const float*     iv = (const float*)d_in[1];      // ivar      [B,1,L]
    const float*     op = (const float*)d_in[2];      // output    [B,1,L]
    const long long* ov = (const long long*)d_in[3];  // overlap_index [B,4] int64

    const int Bn = in_sizes[3] / 4;
    const int L  = in_sizes[0] / Bn;

    float* ws = (float*)d_ws;   // per-sample partials: Bn floats

    chi2_per_sample<<<Bn, THREADS, 0, stream>>>(fl, iv, op, ov, ws, L);
    mean_reduce_wmma<<<1, 32, 0, stream>>>(ws, (float*)d_out, Bn);
}